// YuanExperts_69191923138857
// MI455X (gfx1250) — compile-verified
//
#include <hip/hip_runtime.h>

// Yuan MoE: T=1024, H=2048, E=32, I=1024, top-k=2.
// Sparse gather-per-expert schedule -> HBM-bound on 768MB fp32 weights (~33us @ 23.3TB/s).
// fp32 WMMA 16x16x4 for full reference precision; A panels staged to LDS with
// CDNA5 async global->LDS copies (ASYNCcnt), double-buffered; B (weights) streamed
// straight from global (every 64B line fully consumed; L2 catches M-tile re-reads).

#define T_TOK 1024
#define H_DIM 2048
#define E_EXP 32
#define I_DIM 1024
#define KC    256     // K-chunk staged in LDS per double-buffer slot

typedef __attribute__((ext_vector_type(2))) float v2f;
typedef __attribute__((ext_vector_type(8))) float v8f;

__device__ __forceinline__ v8f wmma_f32(v2f a, v2f b, v8f c) {
    return __builtin_amdgcn_wmma_f32_16x16x4_f32(false, a, false, b, (short)0, c,
                                                 false, false);
}

__device__ __forceinline__ void async_copy16(unsigned lds_addr, const float* gptr) {
    // GLOBAL_LOAD_ASYNC_TO_LDS_B128: per-lane 16B, tracked by ASYNCcnt
    asm volatile("global_load_async_to_lds_b128 %0, %1, off"
                 :: "v"(lds_addr), "v"(gptr) : "memory");
}

// ---------------- Router: one wave32 per token, lane == expert ----------------
__global__ __launch_bounds__(32) void router_kernel(
    const float* __restrict__ x, const float* __restrict__ wqkv,
    int* __restrict__ topk_id, float* __restrict__ topk_w) {
    const int t = blockIdx.x;
    const int l = threadIdx.x;  // expert index, 0..31

    const float4* xr = (const float4*)(x + (size_t)t * H_DIM);
    const float4* wq = (const float4*)(wqkv + (size_t)l * H_DIM);
    const float4* wk = (const float4*)(wqkv + (size_t)(E_EXP + l) * H_DIM);
    const float4* wv = (const float4*)(wqkv + (size_t)(2 * E_EXP + l) * H_DIM);

    float q = 0.f, k = 0.f, v = 0.f;
    for (int h = 0; h < H_DIM / 4; ++h) {
        float4 xv = xr[h];
        float4 a = wq[h]; q += xv.x * a.x + xv.y * a.y + xv.z * a.z + xv.w * a.w;
        float4 b = wk[h]; k += xv.x * b.x + xv.y * b.y + xv.z * b.z + xv.w * b.w;
        float4 c = wv[h]; v += xv.x * c.x + xv.y * c.y + xv.z * c.z + xv.w * c.w;
    }

    __shared__ float ks[E_EXP], vs[E_EXP];
    ks[l] = k; vs[l] = v;
    __syncthreads();

    float mx = -3.4e38f;
    for (int f = 0; f < E_EXP; ++f) mx = fmaxf(mx, q * ks[f]);
    float s = 0.f, acc = 0.f;
    for (int f = 0; f < E_EXP; ++f) {
        float p = __expf(q * ks[f] - mx);
        s += p; acc += p * vs[f];
    }
    float logit = acc / s;

    float v0 = logit; int i0 = l;
    for (int off = 16; off > 0; off >>= 1) {
        float ov = __shfl_xor(v0, off, 32);
        int   oi = __shfl_xor(i0, off, 32);
        if (ov > v0 || (ov == v0 && oi < i0)) { v0 = ov; i0 = oi; }
    }
    float v1 = (l == i0) ? -3.4e38f : logit; int i1 = l;
    for (int off = 16; off > 0; off >>= 1) {
        float ov = __shfl_xor(v1, off, 32);
        int   oi = __shfl_xor(i1, off, 32);
        if (ov > v1 || (ov == v1 && oi < i1)) { v1 = ov; i1 = oi; }
    }
    if (l == 0) {
        float w0 = 1.f / (1.f + __expf(v1 - v0));
        topk_id[t * 2]     = i0;
        topk_id[t * 2 + 1] = i1;
        topk_w[t * 2]      = w0;
        topk_w[t * 2 + 1]  = 1.f - w0;
    }
}

// ------- Deterministic per-expert token list (block e scans tokens in order) -------
__global__ __launch_bounds__(256) void compact_kernel(
    const int* __restrict__ topk_id, const float* __restrict__ topk_w,
    int* __restrict__ counts, int* __restrict__ tlist, float* __restrict__ wlist) {
    const int e = blockIdx.x;
    const int lane = threadIdx.x & 31;
    const int wid  = threadIdx.x >> 5;
    __shared__ int wsum[8];
    __shared__ int base;
    if (threadIdx.x == 0) base = 0;
    __syncthreads();

    for (int t0 = 0; t0 < T_TOK; t0 += 256) {
        int t = t0 + threadIdx.x;
        int id0 = topk_id[t * 2], id1 = topk_id[t * 2 + 1];
        int sel = (id0 == e) || (id1 == e);
        unsigned long long bal = __ballot(sel);
        int lanepre = __popcll(bal & ((1ull << lane) - 1ull));
        if (lane == 0) wsum[wid] = __popcll(bal);
        __syncthreads();
        int wpre = 0, tot = 0;
        for (int j = 0; j < 8; ++j) { if (j < wid) wpre += wsum[j]; tot += wsum[j]; }
        if (sel) {
            int pos = base + wpre + lanepre;
            tlist[e * T_TOK + pos] = t;
            wlist[e * T_TOK + pos] = (id0 == e) ? topk_w[t * 2] : topk_w[t * 2 + 1];
        }
        __syncthreads();
        if (threadIdx.x == 0) base += tot;
        __syncthreads();
    }
    if (threadIdx.x == 0) counts[e] = base;
}

__global__ void offsets_kernel(const int* __restrict__ counts, int* __restrict__ offs) {
    if (threadIdx.x == 0) {
        int s = 0;
        for (int e = 0; e < E_EXP; ++e) { offs[e] = s; s += counts[e]; }
        offs[E_EXP] = s;
    }
}

// ------- Gate/Up GEMM + SiLU + combine, per (expert, 16-token tile, 128 I-cols) -------
// A panel (16 gathered x rows) streamed into LDS via async global->LDS, double-buffered.
__global__ __launch_bounds__(256) void gateup_kernel(
    const float* __restrict__ x, const float* __restrict__ w1,
    const int* __restrict__ counts, const int* __restrict__ offs,
    const int* __restrict__ tlist, const float* __restrict__ wlist,
    float* __restrict__ act) {
    const int e    = blockIdx.z;
    const int tile = blockIdx.x;
    const int cnt  = counts[e];
    if (tile * 16 >= cnt) return;  // uniform: EXEC stays all-1s for WMMA

    __shared__ int   tok[16];
    __shared__ float twt[16];
    __shared__ float sA[2][16][KC + 4];  // +4 pad: conflict-free ds_load_b64 fragments

    if (threadIdx.x < 16) {
        int j  = tile * 16 + threadIdx.x;
        int jc = j < cnt ? j : cnt - 1;
        tok[threadIdx.x] = tlist[e * T_TOK + jc];
        twt[threadIdx.x] = wlist[e * T_TOK + jc];
    }
    __syncthreads();

    const int tid  = threadIdx.x;
    const int srow = tid >> 4;          // staging: row 0..15
    const int scol = (tid & 15) * 16;   // staging: 16-float column group
    const float* srcrow = x + (size_t)tok[srow] * H_DIM + scol;

    const int l    = threadIdx.x & 31;
    const int wv   = threadIdx.x >> 5;
    const int half = l >> 4;
    const int mn   = l & 15;
    const int koff = 2 * half;

    const int ct = blockIdx.y * 8 + wv;  // this wave's gate-column tile (0..63)
    const float* w1g = w1 + ((size_t)e * 2 * I_DIM + (size_t)ct * 16 + mn) * H_DIM + koff;
    const float* w1u = w1g + (size_t)I_DIM * H_DIM;

    // stage chunk 0
    {
        unsigned lds0 = (unsigned)(size_t)&sA[0][srow][scol];
#pragma unroll
        for (int qd = 0; qd < 4; ++qd) async_copy16(lds0 + qd * 16, srcrow + qd * 4);
    }

    v8f accg = {}, accu = {};
    const int nchunk = H_DIM / KC;  // 8
    for (int kb = 0; kb < nchunk; ++kb) {
        if (kb + 1 < nchunk) {
            unsigned lds0 = (unsigned)(size_t)&sA[(kb + 1) & 1][srow][scol];
            const float* g = srcrow + (kb + 1) * KC;
#pragma unroll
            for (int qd = 0; qd < 4; ++qd) async_copy16(lds0 + qd * 16, g + qd * 4);
            // previous chunk's 4 copies complete (in-order), newest 4 may fly
            asm volatile("s_wait_asynccnt 0x4" ::: "memory");
        } else {
            asm volatile("s_wait_asynccnt 0x0" ::: "memory");
        }
        __syncthreads();

        const float* abase = &sA[kb & 1][mn][koff];
        const float* bg = w1g + kb * KC;
        const float* bu = w1u + kb * KC;
#pragma unroll 4
        for (int k4 = 0; k4 < KC; k4 += 4) {
            v2f a  = *(const v2f*)(abase + k4);  // ds_load_b64
            v2f vg = *(const v2f*)(bg + k4);
            v2f vu = *(const v2f*)(bu + k4);
            accg = wmma_f32(a, vg, accg);
            accu = wmma_f32(a, vu, accu);
        }
        __syncthreads();  // readers done before this buffer is re-staged
    }

    const size_t slotbase = (size_t)offs[e] + (size_t)tile * 16;
#pragma unroll
    for (int r = 0; r < 8; ++r) {
        int m = r + 8 * half;
        int j = tile * 16 + m;
        if (j < cnt) {
            float g = accg[r], u = accu[r];
            float hval = g / (1.f + __expf(-g)) * u * twt[m];  // silu(g)*u*combine
            act[(slotbase + m) * I_DIM + ct * 16 + mn] = hval;
        }
    }
}

// ---- Down projection: act @ w2^T, A staged async to LDS, atomic-add into out ----
__global__ __launch_bounds__(256) void down_kernel(
    const float* __restrict__ act, const float* __restrict__ w2,
    const int* __restrict__ counts, const int* __restrict__ offs,
    const int* __restrict__ tlist, float* __restrict__ out) {
    const int e    = blockIdx.z;
    const int tile = blockIdx.x;
    const int cnt  = counts[e];
    if (tile * 16 >= cnt) return;

    __shared__ int   tok[16];
    __shared__ float sA[2][16][KC + 4];
    if (threadIdx.x < 16) {
        int j = tile * 16 + threadIdx.x;
        tok[threadIdx.x] = tlist[e * T_TOK + (j < cnt ? j : cnt - 1)];
    }
    __syncthreads();

    const int tid  = threadIdx.x;
    const int srow = tid >> 4;
    const int scol = (tid & 15) * 16;
    const size_t slot = (size_t)offs[e] + (size_t)tile * 16;
    const float* srcrow = act + (slot + srow) * I_DIM + scol;  // rows >= cnt: garbage, never stored

    const int l    = threadIdx.x & 31;
    const int wv   = threadIdx.x >> 5;
    const int half = l >> 4;
    const int mn   = l & 15;
    const int koff = 2 * half;

    const int ht0 = blockIdx.y * 16 + wv;      // this wave's two h-col tiles
    const int ht1 = ht0 + 8;
    const float* w2r0 = w2 + ((size_t)e * H_DIM + (size_t)ht0 * 16 + mn) * I_DIM + koff;
    const float* w2r1 = w2 + ((size_t)e * H_DIM + (size_t)ht1 * 16 + mn) * I_DIM + koff;

    {
        unsigned lds0 = (unsigned)(size_t)&sA[0][srow][scol];
#pragma unroll
        for (int qd = 0; qd < 4; ++qd) async_copy16(lds0 + qd * 16, srcrow + qd * 4);
    }

    v8f acc0 = {}, acc1 = {};
    const int nchunk = I_DIM / KC;  // 4
    for (int kb = 0; kb < nchunk; ++kb) {
        if (kb + 1 < nchunk) {
            unsigned lds0 = (unsigned)(size_t)&sA[(kb + 1) & 1][srow][scol];
            const float* g = srcrow + (kb + 1) * KC;
#pragma unroll
            for (int qd = 0; qd < 4; ++qd) async_copy16(lds0 + qd * 16, g + qd * 4);
            asm volatile("s_wait_asynccnt 0x4" ::: "memory");
        } else {
            asm volatile("s_wait_asynccnt 0x0" ::: "memory");
        }
        __syncthreads();

        const float* abase = &sA[kb & 1][mn][koff];
        const float* b0 = w2r0 + kb * KC;
        const float* b1 = w2r1 + kb * KC;
#pragma unroll 4
        for (int k4 = 0; k4 < KC; k4 += 4) {
            v2f a  = *(const v2f*)(abase + k4);
            v2f vb0 = *(const v2f*)(b0 + k4);
            v2f vb1 = *(const v2f*)(b1 + k4);
            acc0 = wmma_f32(a, vb0, acc0);
            acc1 = wmma_f32(a, vb1, acc1);
        }
        __syncthreads();
    }

#pragma unroll
    for (int r = 0; r < 8; ++r) {
        int m = r + 8 * half;
        int j = tile * 16 + m;
        if (j < cnt) {
            atomicAdd(&out[(size_t)tok[m] * H_DIM + ht0 * 16 + mn], acc0[r]);
            atomicAdd(&out[(size_t)tok[m] * H_DIM + ht1 * 16 + mn], acc1[r]);
        }
    }
}

extern "C" void kernel_launch(void* const* d_in, const int* in_sizes, int n_in,
                              void* d_out, int out_size, void* d_ws, size_t ws_size,
                              hipStream_t stream) {
    const float* x    = (const float*)d_in[0];  // [T, H]
    const float* wqkv = (const float*)d_in[1];  // [3E, H]
    const float* w1   = (const float*)d_in[2];  // [E, 2I, H]
    const float* w2   = (const float*)d_in[3];  // [E, H, I]
    float* out = (float*)d_out;                  // [T, H]

    char* ws = (char*)d_ws;
    int*   topk_id = (int*)(ws + 0);                 // 2048 ints
    float* topk_w  = (float*)(ws + 8192);            // 2048 floats
    int*   counts  = (int*)(ws + 16384);             // 32 ints
    int*   offs    = (int*)(ws + 16640);             // 33 ints
    int*   tlist   = (int*)(ws + 17152);             // E*T ints
    float* wlist   = (float*)(ws + 17152 + 131072);  // E*T floats
    float* act     = (float*)(ws + 17152 + 262144);  // (T*2 + 16) * I floats (~8.3 MB)

    hipMemsetAsync(d_out, 0, (size_t)out_size * sizeof(float), stream);

    router_kernel<<<T_TOK, 32, 0, stream>>>(x, wqkv, topk_id, topk_w);
    compact_kernel<<<E_EXP, 256, 0, stream>>>(topk_id, topk_w, counts, tlist, wlist);
    offsets_kernel<<<1, 32, 0, stream>>>(counts, offs);

    dim3 ggrid(T_TOK / 16, I_DIM / 128, E_EXP);   // (M-tile, 128-col block, expert)
    gateup_kernel<<<ggrid, 256, 0, stream>>>(x, w1, counts, offs, tlist, wlist, act);

    dim3 dgrid(T_TOK / 16, H_DIM / 256, E_EXP);   // (M-tile, 256-col block, expert)
    down_kernel<<<dgrid, 256, 0, stream>>>(act, w2, counts, offs, tlist, out);
}